// RevRNN_81905026334841
// MI455X (gfx1250) — compile-verified
//
#include <hip/hip_runtime.h>
#include <cmath>

// ---------------------------------------------------------------------------
// RevRNN scan for MI455X (gfx1250): persistent 16-block cooperative kernel.
//  - fp32 WMMA (v_wmma_f32_16x16x4_f32) GEMM core, A staged in LDS
//  - Tensor Data Mover (tensor_load_to_lds) stages the per-step x slice with
//    hardware LDS padding (256 dwords data + 4 dwords pad = 260-float rows)
//  - monotonic atomic ticket barrier for the 1040 device-wide syncs
// ---------------------------------------------------------------------------

typedef float        v2f  __attribute__((ext_vector_type(2)));
typedef float        v8f  __attribute__((ext_vector_type(8)));
typedef unsigned int u32x4 __attribute__((ext_vector_type(4)));
typedef int          i32x4 __attribute__((ext_vector_type(4)));
typedef int          i32x8 __attribute__((ext_vector_type(8)));

#define NBLK   16
#define NTHR   256
#define Bsz    64
#define Ssz    512
#define Fsz    256
#define Hsz    512       // half hidden
#define HID2   1024      // 2H
#define Tsteps 520       // S + DELAY
#define KDIM   770       // F + 2 + H
#define AH_STRIDE 516    // 512 h-cols + 4 pad floats (bank skew, never read)
#define AS_STRIDE 260    // 256 x + pos0 + pos1 + 2 zero floats
#define LDS_FLOATS (Bsz*AH_STRIDE + Bsz*AS_STRIDE)   // 49,664 floats = 198,656 B

#if __has_builtin(__builtin_amdgcn_tensor_load_to_lds)
#define HAVE_TDM 1
#else
#define HAVE_TDM 0
#endif

#define WMMA_F32(acc, a, b)                                                   \
    acc = __builtin_amdgcn_wmma_f32_16x16x4_f32(false, (a), false, (b),       \
                                                (short)0, (acc), false, false)

__device__ __forceinline__ void grid_barrier(unsigned* ctr, unsigned target) {
    __syncthreads();
    if (threadIdx.x == 0) {
        __threadfence();                       // publish our state writes
        atomicAdd(ctr, 1u);
        while (__hip_atomic_load(ctr, __ATOMIC_ACQUIRE, __HIP_MEMORY_SCOPE_AGENT) < target) {
            __builtin_amdgcn_s_sleep(2);
        }
        __threadfence();                       // observe everyone else's writes
    }
    __syncthreads();
}

__global__ void __launch_bounds__(NTHR, 1)
revrnn_kernel(const float* __restrict__ x,       // (64, 512, 256)
              const float* __restrict__ h0,      // (1024,)
              const float* __restrict__ p0,      // (1024, 770) row-major
              const float* __restrict__ p1,      // (1024, 770)
              float* __restrict__ state,         // (64, 1024) == d_out
              unsigned* __restrict__ ctr)
{
    extern __shared__ float smem[];
    float* Ah = smem;                          // [64][AH_STRIDE] centered h-half
    float* As = smem + Bsz * AH_STRIDE;        // [64][AS_STRIDE] x|pos|0|0

    const int tid  = threadIdx.x;
    const int g    = blockIdx.x;
    const int lane = tid & 31;
    const int w    = tid >> 5;                 // wave id 0..7
    const int lo   = lane & 15;
    const int hi   = lane >> 4;                // 0 | 1

    unsigned epoch = 0;

    // ---- init: h = broadcast(hidden_state); zero As cols 258,259 once ------
    for (int i = g * NTHR + tid; i < Bsz * HID2; i += NBLK * NTHR)
        state[i] = h0[i & (HID2 - 1)];
    if (tid < 128)
        As[(tid >> 1) * AS_STRIDE + 258 + (tid & 1)] = 0.0f;
    epoch++; grid_barrier(ctr, epoch * NBLK);

    // ---- tile assignment ---------------------------------------------------
    const int m0  = (w & 3) * 16;              // M tile base (batch rows)
    const int ip  = w >> 2;                    // 16-col group (0|1)
    const int ca0 = g * 32 + ip * 16;          // h-column base in [0,512)
    const int na0 = ca0;                       // 'a' columns of z
    const int nb0 = Hsz + ca0;                 // 'b' columns of z

    const float* arowH = &Ah[(m0 + lo) * AH_STRIDE + 2 * hi];
    const float* arowS = &As[(m0 + lo) * AS_STRIDE + 2 * hi];
#if HAVE_TDM
    const unsigned asBase = (unsigned)(size_t)(void*)As;   // LDS byte offset
#endif

    for (int t = 0; t < Tsteps; ++t) {
        // ---- stage x slice into As cols [0,256) -----------------------------
#if HAVE_TDM
        if (w == 0 && t < Ssz) {
            // D# group0: count=1 | lds_addr | 57-bit global addr | type=2
            const unsigned long long ga =
                (unsigned long long)(const void*)x + (unsigned long long)t * (Fsz * 4);
            u32x4 g0 = { 1u, asBase, (unsigned)ga,
                         (unsigned)((ga >> 32) & 0x01FFFFFFu) | 0x80000000u };
            // D# group1: data_size=4B, pad_enable, pad every 256 dwords by 4 dwords,
            // tensor_dim0=256, tensor_dim1=64, tile=256x64, dim0_stride=131072
            i32x8 g1 = { (int)((2u << 16) | (1u << 20) | (7u << 22) | (3u << 25)),
                         (int)(256u << 16),      // tensor_dim0[15:0] << 16
                         (int)(64u  << 16),      // tensor_dim1[15:0] << 16
                         (int)(256u << 16),      // tile_dim0 << 16
                         64,                     // tile_dim1
                         0x20000,                // tensor_dim0_stride = 131072
                         0, 0 };
            i32x4 gz4 = { 0, 0, 0, 0 };
            i32x8 gz8 = { 0, 0, 0, 0, 0, 0, 0, 0 };
            __builtin_amdgcn_tensor_load_to_lds(g0, g1, gz4, gz4, gz8, 0);
        }
#else
        if (t < Ssz) {
            for (int idx = tid; idx < Bsz * Fsz; idx += NTHR) {
                const int r = idx >> 8, c = idx & 255;
                As[r * AS_STRIDE + c] =
                    x[(size_t)r * (Ssz * Fsz) + (size_t)t * Fsz + c];
            }
        }
#endif
        if (t == Ssz) {   // delay region: x contribution becomes zero
            for (int idx = tid; idx < Bsz * Fsz; idx += NTHR)
                As[(idx >> 8) * AS_STRIDE + (idx & 255)] = 0.0f;
        }
        // pos features for this step
        if (tid < 128) {
            const float pos0 = (float)(t + 1);
            const float pos1 = ((float)(t + 1) - 260.5f) / 260.5f;  // (T+1)/2
            As[(tid >> 1) * AS_STRIDE + 256 + (tid & 1)] = (tid & 1) ? pos1 : pos0;
        }

        for (int phase = 0; phase < 2; ++phase) {
            const int    srcOff = (phase == 0) ? 0 : Hsz;   // GEMM input half
            const int    rwOff  = (phase == 0) ? Hsz : 0;   // residual+dest half
            const float* Wp     = (phase == 0) ? p0 : p1;

            // ---- stage batch-mean-centered h-half into Ah (overlaps TDM) ---
            for (int k = tid; k < Hsz; k += NTHR) {
                float sum = 0.f;
                for (int r = 0; r < Bsz; ++r) sum += state[r * HID2 + srcOff + k];
                const float mu = sum * (1.0f / 64.0f);
                for (int r = 0; r < Bsz; ++r)
                    Ah[r * AH_STRIDE + k] = state[r * HID2 + srcOff + k] - mu;
            }
#if HAVE_TDM
            if (phase == 0 && w == 0 && t < Ssz)
                __builtin_amdgcn_s_wait_tensorcnt(0);       // TDM -> LDS done
#endif
            __syncthreads();

            // ---- z tiles via fp32 WMMA: D = A(16x4) x B(4x16) + C ----------
            v8f acc_a = {0.f, 0.f, 0.f, 0.f, 0.f, 0.f, 0.f, 0.f};
            v8f acc_b = {0.f, 0.f, 0.f, 0.f, 0.f, 0.f, 0.f, 0.f};
            const float* wa = Wp + (size_t)(na0 + lo) * KDIM + 2 * hi;
            const float* wb = Wp + (size_t)(nb0 + lo) * KDIM + 2 * hi;

            #pragma unroll 4
            for (int k = 0; k < Hsz; k += 4) {              // K segment 1: h-half
                v2f a  = *(const v2f*)(arowH + k);
                v2f ba = *(const v2f*)(wa + k);
                v2f bb = *(const v2f*)(wb + k);
                WMMA_F32(acc_a, a, ba);
                WMMA_F32(acc_b, a, bb);
            }
            const float* was = wa + Hsz;
            const float* wbs = wb + Hsz;
            #pragma unroll 4
            for (int k = 0; k < 256; k += 4) {              // K segment 2: x part
                v2f a  = *(const v2f*)(arowS + k);
                v2f ba = *(const v2f*)(was + k);
                v2f bb = *(const v2f*)(wbs + k);
                WMMA_F32(acc_a, a, ba);
                WMMA_F32(acc_b, a, bb);
            }
            {   // K tail: As cols 256..259 = pos0,pos1,0,0 ; keep W in-bounds
                v2f a  = *(const v2f*)(arowS + 256);
                v2f ta = *(const v2f*)(was + 256 - 2 * hi);
                v2f tb = *(const v2f*)(wbs + 256 - 2 * hi);
                v2f zz = {0.f, 0.f};
                v2f ba = hi ? zz : ta;
                v2f bb = hi ? zz : tb;
                WMMA_F32(acc_a, a, ba);
                WMMA_F32(acc_b, a, bb);
            }

            // ---- h = sigmoid(za)*tanh(zb); reversible residual, in place ---
            #pragma unroll
            for (int r = 0; r < 8; ++r) {
                const float za = acc_a[r];
                const float zb = acc_b[r];
                const float hc = (1.0f / (1.0f + __expf(-za))) * tanhf(zb);
                const int row = m0 + r + 8 * hi;           // ISA C/D layout
                const int col = ca0 + lo;
                const int off = row * HID2 + rwOff + col;
                state[off] = state[off] + hc;
            }
            epoch++; grid_barrier(ctr, epoch * NBLK);
        }
    }
    // state == d_out already holds h_final = [y0 | y1]
}

extern "C" void kernel_launch(void* const* d_in, const int* in_sizes, int n_in,
                              void* d_out, int out_size, void* d_ws, size_t ws_size,
                              hipStream_t stream) {
    (void)in_sizes; (void)n_in; (void)out_size; (void)ws_size;
    const float* x  = (const float*)d_in[0];
    const float* h0 = (const float*)d_in[1];
    const float* p0 = (const float*)d_in[2];
    const float* p1 = (const float*)d_in[3];
    float*    state = (float*)d_out;
    unsigned*   ctr = (unsigned*)d_ws;

    const size_t lds_bytes = (size_t)LDS_FLOATS * sizeof(float);  // 198,656 B < 320 KB
    (void)hipFuncSetAttribute((const void*)revrnn_kernel,
                              hipFuncAttributeMaxDynamicSharedMemorySize, (int)lds_bytes);
    (void)hipMemsetAsync(d_ws, 0, 256, stream);   // reset grid-barrier ticket counter
    revrnn_kernel<<<dim3(NBLK), dim3(NTHR), lds_bytes, stream>>>(x, h0, p0, p1, state, ctr);
}